// IGNN_60026462929132
// MI455X (gfx1250) — compile-verified
//
#include <hip/hip_runtime.h>
#include <math.h>

// IGNN on MI455X (gfx1250, wave32).
// - Dense GEMMs: V_WMMA_F32_16X16X4_F32; LDS-staged swizzled B operand with
//   8-fragment preload per K-step so DS waits pipeline against WMMA issue.
// - SpMM (dominant cost): CSR gather, wave-per-node, 32-edge batches with
//   __shfl broadcast of (row,val). No f32 atomics in the 42-SpMM hot loop.

typedef float v2f __attribute__((ext_vector_type(2)));
typedef float v8f __attribute__((ext_vector_type(8)));

#define IG_KAPPA 0.9f
#define IG_M 128
#define IG_P 256
#define IG_MITR 20
#define IG_POWER 100

// ------------------------- CSR construction --------------------------------
__global__ void ignn_hist(const int* __restrict__ key, int* __restrict__ cnt, int E) {
  int e = blockIdx.x * blockDim.x + threadIdx.x;
  if (e < E) atomicAdd(&cnt[key[e]], 1);
}

// Single-block tiled exclusive scan: out[0..n] (out[n] = total).
__global__ void ignn_exscan(const int* __restrict__ in, int* __restrict__ out, int n) {
  __shared__ int sh[256];
  __shared__ int carry_s;
  if (threadIdx.x == 0) carry_s = 0;
  __syncthreads();
  for (int base = 0; base < n; base += 256) {
    int i = base + threadIdx.x;
    int x = (i < n) ? in[i] : 0;
    sh[threadIdx.x] = x;
    __syncthreads();
    for (int off = 1; off < 256; off <<= 1) {
      int add = (threadIdx.x >= off) ? sh[threadIdx.x - off] : 0;
      __syncthreads();
      sh[threadIdx.x] += add;
      __syncthreads();
    }
    int incl = sh[threadIdx.x];
    int carry = carry_s;
    if (i < n) out[i] = carry + incl - x;
    __syncthreads();
    if (threadIdx.x == 255) carry_s = carry + sh[255];
    __syncthreads();
  }
  if (threadIdx.x == 0) out[n] = carry_s;
}

__global__ void ignn_csr_scatter(const int* __restrict__ key, const int* __restrict__ other,
                                 const float* __restrict__ vals, int* __restrict__ cursor,
                                 int* __restrict__ sother, float* __restrict__ svals, int E) {
  int e = blockIdx.x * blockDim.x + threadIdx.x;
  if (e < E) {
    int pos = atomicAdd(&cursor[key[e]], 1);
    sother[pos] = other[e];
    svals[pos] = vals[e];
  }
}

// ------------------------- spectral radius ---------------------------------
__global__ void ignn_init_v(float* __restrict__ v, int N) {
  int i = blockIdx.x * blockDim.x + threadIdx.x;
  if (i < N) v[i] = rsqrtf((float)N);
}

// w[r] = sum_{e in row r} val[e] * v[col[e]]   (row-CSR, thread per row)
__global__ void ignn_spmv_csr(const int* __restrict__ rowptr, const int* __restrict__ scols,
                              const float* __restrict__ svals, const float* __restrict__ v,
                              float* __restrict__ w, int N) {
  int r = blockIdx.x * blockDim.x + threadIdx.x;
  if (r >= N) return;
  float acc = 0.0f;
  int e0 = rowptr[r], e1 = rowptr[r + 1];
  for (int i = e0; i < e1; ++i) acc += svals[i] * v[scols[i]];
  w[r] = acc;
}

__global__ void ignn_sumsq(const float* __restrict__ w, float* __restrict__ ss, int N) {
  __shared__ float red[256];
  int i = blockIdx.x * blockDim.x + threadIdx.x;
  float x = (i < N) ? w[i] : 0.0f;
  red[threadIdx.x] = x * x;
  __syncthreads();
  for (int s = 128; s > 0; s >>= 1) {
    if (threadIdx.x < s) red[threadIdx.x] += red[threadIdx.x + s];
    __syncthreads();
  }
  if (threadIdx.x == 0) atomicAdd(ss, red[0]);
}

__global__ void ignn_normalize(const float* __restrict__ w, float* __restrict__ v,
                               const float* __restrict__ ss, float* __restrict__ rho, int N) {
  int i = blockIdx.x * blockDim.x + threadIdx.x;
  const float r = sqrtf(ss[0]);
  if (i == 0) *rho = r;
  if (i < N) v[i] = w[i] / (r + 1e-12f);
}

// ----------------- proj_norm_inf (one block per row of W) ------------------
__global__ void ignn_proj_norm_inf(const float* __restrict__ W,
                                   const float* __restrict__ rho_p,
                                   float* __restrict__ WpT) {
  const int j = blockIdx.x;   // row 0..127
  const int t = threadIdx.x;  // 0..127
  const float vmax = IG_KAPPA / rho_p[0];
  __shared__ float u[128];
  __shared__ float cs[128];
  __shared__ int rc;
  __shared__ float stheta, stotal;

  float wv = W[j * IG_M + t];
  float av = fabsf(wv);
  u[t] = av;
  __syncthreads();
  for (int k = 2; k <= 128; k <<= 1) {        // ascending bitonic sort
    for (int s = k >> 1; s > 0; s >>= 1) {
      int p = t ^ s;
      if (p > t) {
        float a = u[t], b = u[p];
        bool up = ((t & k) == 0);
        if ((up && a > b) || (!up && a < b)) { u[t] = b; u[p] = a; }
      }
      __syncthreads();
    }
  }
  float ud = u[127 - t];                       // descending
  cs[t] = ud;
  __syncthreads();
  for (int off = 1; off < 128; off <<= 1) {    // inclusive prefix sum
    float add = (t >= off) ? cs[t - off] : 0.0f;
    __syncthreads();
    cs[t] += add;
    __syncthreads();
  }
  float cssv = cs[t] - vmax;
  if (t == 0) rc = 0;
  __syncthreads();
  if (ud * (float)(t + 1) > cssv) atomicAdd(&rc, 1);
  __syncthreads();
  if (t == 0) {
    int r = rc > 1 ? rc : 1;
    stheta = (cs[r - 1] - vmax) / (float)r;
    stotal = cs[127];
  }
  __syncthreads();
  float outv;
  if (stotal > vmax) {
    float p = fmaxf(av - stheta, 0.0f);
    outv = (wv > 0.0f) ? p : ((wv < 0.0f) ? -p : 0.0f);
  } else {
    outv = wv;
  }
  WpT[t * IG_M + j] = outv;   // transposed for GEMM B operand
}

// --------------------------- transpose -------------------------------------
__global__ void ignn_transpose(const float* __restrict__ In, float* __restrict__ Out,
                               int R, int C) {
  size_t idx = (size_t)blockIdx.x * blockDim.x + threadIdx.x;
  if (idx < (size_t)R * C) {
    size_t r = idx / C, c = idx % C;
    Out[c * (size_t)R + r] = In[idx];
  }
}

// --------------------- dense GEMM via f32 WMMA -----------------------------
// C[N,128] = A[N,K] @ Bt[K,128], row-major f32.  One wave: 16-row strip x all
// 8 column tiles.  B staged per 64-row K-chunk into LDS, swizzled
// Bs[q][col][kk] = Bt[4q+kk][col] so each lane's (k,k+1) pair is one
// 8B-aligned conflict-free ds_load_b64.  All 8 B fragments of a K-step are
// preloaded before the first WMMA so s_wait_dscnt forms a pipeline ladder.
// Fragment layouts per CDNA5 ISA 7.12.2 (A 16x4, B 4x16, C/D 8 VGPR).
__global__ void ignn_gemm_wmma(const float* __restrict__ A, const float* __restrict__ Bt,
                               float* __restrict__ C, int Nrows, int K) {
  __shared__ float Bs[16 * 128 * 4];  // 32 KB: one 64-row K-chunk
  const int lane = threadIdx.x & 31;
  const int wave = threadIdx.x >> 5;
  const int tn = blockIdx.x * (blockDim.x >> 5) + wave;
  const int ntiles = (Nrows + 15) >> 4;
  const bool active = (tn < ntiles);  // wave-uniform; all waves keep hitting barriers

  const int m = lane & 15;
  const int kb = (lane >> 4) << 1;             // 0 or 2
  int row = tn * 16 + m;
  if (row > Nrows - 1) row = Nrows - 1;        // clamp: branch-free, never stored OOB

  v8f acc[8] = {};
  for (int kc = 0; kc < K; kc += 64) {
    // cooperative stage: 64x128 floats, fixed 8 float4-iters per thread
#pragma unroll
    for (int it2 = 0; it2 < 8; ++it2) {
      int idx = (int)threadIdx.x + it2 * 256;  // 0..2047
      int k = idx >> 5;                        // 0..63
      int c4 = (idx & 31) << 2;                // col base
      float4 bv = *(const float4*)(Bt + (size_t)(kc + k) * IG_M + c4);
      int q = k >> 2, kk = k & 3;
      Bs[(q * 128 + c4 + 0) * 4 + kk] = bv.x;
      Bs[(q * 128 + c4 + 1) * 4 + kk] = bv.y;
      Bs[(q * 128 + c4 + 2) * 4 + kk] = bv.z;
      Bs[(q * 128 + c4 + 3) * 4 + kk] = bv.w;
    }
    __syncthreads();
    if (active) {
      for (int k0 = 0; k0 < 64; k0 += 4) {
        const int q = k0 >> 2;
        const float* ap = A + (size_t)row * K + kc + k0 + kb;
        v2f a;
        a.x = ap[0];
        a.y = ap[1];
        v2f bfr[8];
#pragma unroll
        for (int t = 0; t < 8; ++t)
          bfr[t] = *(const v2f*)&Bs[(q * 128 + t * 16 + m) * 4 + kb];
#pragma unroll
        for (int t = 0; t < 8; ++t)
          acc[t] = __builtin_amdgcn_wmma_f32_16x16x4_f32(
              false, a, false, bfr[t], (short)0, acc[t], false, false);
      }
    }
    __syncthreads();
  }
  if (active) {
    const int rbase = tn * 16 + ((lane >> 4) << 3);
#pragma unroll
    for (int t = 0; t < 8; ++t) {
#pragma unroll
      for (int r8 = 0; r8 < 8; ++r8) {
        int orow = rbase + r8;
        if (orow < Nrows) C[(size_t)orow * IG_M + t * 16 + m] = acc[t][r8];
      }
    }
  }
}

// ------------------------------- SpMM (CSR gather) -------------------------
// Out[c,:] = sum_{e in col-segment c} val[e] * G[row[e],:].  One wave per
// node.  Edge metadata is loaded coalesced in 32-edge batches and broadcast
// with __shfl (wave32): ~1 VMEM instr/edge + one 512B coalesced gather/edge,
// register accumulate, single 512B store.  No atomics, no output zeroing.
__global__ void ignn_spmm_csr(const int* __restrict__ colptr, const int* __restrict__ srows,
                              const float* __restrict__ svals, const float* __restrict__ G,
                              float* __restrict__ Out, int N) {
  int c = blockIdx.x * (blockDim.x >> 5) + (threadIdx.x >> 5);
  if (c >= N) return;
  int lane = threadIdx.x & 31;
  int e0 = colptr[c], e1 = colptr[c + 1];
  float4 acc = {0.0f, 0.0f, 0.0f, 0.0f};
  for (int base = e0; base < e1; base += 32) {
    int me = base + lane;
    int r_l = (me < e1) ? srows[me] : 0;       // coalesced batch load
    float v_l = (me < e1) ? svals[me] : 0.0f;  // coalesced batch load
    int nb = e1 - base;
    if (nb > 32) nb = 32;
    for (int j = 0; j < nb; ++j) {
      int r = __shfl(r_l, j);                  // wave32 broadcast
      float v = __shfl(v_l, j);
      const float4 g = *(const float4*)(G + (size_t)r * IG_M + lane * 4);
      acc.x += v * g.x;
      acc.y += v * g.y;
      acc.z += v * g.z;
      acc.w += v * g.w;
    }
  }
  *(float4*)(Out + (size_t)c * IG_M + lane * 4) = acc;
}

// ------------------------------ elementwise --------------------------------
__global__ void ignn_relu_add(const float* __restrict__ Acc, const float* __restrict__ Bm,
                              float* __restrict__ X, size_t n) {
  size_t i = (size_t)blockIdx.x * blockDim.x + threadIdx.x;
  if (i < n) X[i] = fmaxf(Acc[i] + Bm[i], 0.0f);
}

__global__ void ignn_elu_combine(const float* __restrict__ H, const float* __restrict__ T,
                                 const float* __restrict__ bias, float* __restrict__ X2,
                                 size_t n) {
  size_t i = (size_t)blockIdx.x * blockDim.x + threadIdx.x;
  if (i < n) {
    float x = H[i] + T[i] + bias[i & (IG_M - 1)];
    X2[i] = (x > 0.0f) ? x : expm1f(x);
  }
}

__global__ void ignn_final_combine(const float* __restrict__ H, const float* __restrict__ T,
                                   const float* __restrict__ bias, float* __restrict__ out,
                                   size_t n) {
  size_t i = (size_t)blockIdx.x * blockDim.x + threadIdx.x;
  if (i < n) out[i] = H[i] + T[i] + bias[i & (IG_M - 1)];
}

// ------------------------------- launcher ----------------------------------
extern "C" void kernel_launch(void* const* d_in, const int* in_sizes, int n_in,
                              void* d_out, int out_size, void* d_ws, size_t ws_size,
                              hipStream_t stream) {
  (void)n_in; (void)out_size; (void)ws_size;
  const float* features = (const float*)d_in[0];   // [P, N]
  const float* edge_vals = (const float*)d_in[1];  // [E]
  const float* W1 = (const float*)d_in[2];
  const float* Omega1 = (const float*)d_in[3];
  const float* W2 = (const float*)d_in[4];
  const float* Omega2 = (const float*)d_in[5];
  const float* V0_w = (const float*)d_in[6];
  const float* V0_b = (const float*)d_in[7];
  const float* V1_w = (const float*)d_in[8];
  const float* V1_b = (const float*)d_in[9];
  const int* erows = (const int*)d_in[10];
  const int* ecols = (const int*)d_in[11];

  const int N = in_sizes[0] / IG_P;  // 50000
  const int E = in_sizes[1];         // 800000
  const size_t NM = (size_t)N * IG_M;

  // ---- workspace bump allocator (256B aligned regions) ----
  char* wsb = (char*)d_ws;
  size_t off = 0;
  auto alloc = [&](size_t nelem) -> void* {
    void* p = (void*)(wsb + off);
    off += ((nelem * 4) + 255) & ~(size_t)255;
    return p;
  };
  float* featT = (float*)alloc((size_t)N * IG_P);  // [N,256]; reused as X2
  float* Bm   = (float*)alloc(NM);
  float* X    = (float*)alloc(NM);
  float* G    = (float*)alloc(NM);
  float* Acc  = (float*)alloc(NM);
  float* vvec = (float*)alloc(N);
  float* wvec = (float*)alloc(N);
  float* scal = (float*)alloc(64);  // [0]=sumsq, [1]=rho
  float* OT1  = (float*)alloc((size_t)IG_P * IG_M);
  float* V0T  = (float*)alloc((size_t)IG_P * IG_M);
  float* OT2  = (float*)alloc((size_t)IG_M * IG_M);
  float* V1T  = (float*)alloc((size_t)IG_M * IG_M);
  float* Wp1T = (float*)alloc((size_t)IG_M * IG_M);
  float* Wp2T = (float*)alloc((size_t)IG_M * IG_M);
  int* cntA    = (int*)alloc(N);        // histogram / cursor scratch
  int* colptr  = (int*)alloc(N + 1);
  int* rowptr  = (int*)alloc(N + 1);
  int* srowsC  = (int*)alloc(E);        // col-sorted: source rows
  float* svalsC = (float*)alloc(E);
  int* scolsR  = (int*)alloc(E);        // row-sorted: source cols
  float* svalsR = (float*)alloc(E);
  float* ss = scal;
  float* rho = scal + 1;

  const int eltBlocks = (int)((NM + 255) / 256);
  const int nBlocks = (N + 255) / 256;
  const int eBlocks = (E + 255) / 256;
  const int nodeWaveBlocks = (N + 7) / 8;  // 8 waves/block, wave per node
  const int gemmBlocks = (((N + 15) / 16) + 7) / 8;

  auto gemm = [&](const float* A, const float* Bt, float* C, int K) {
    ignn_gemm_wmma<<<gemmBlocks, 256, 0, stream>>>(A, Bt, C, N, K);
  };
  auto spmm = [&](const float* Gm, float* Out) {
    ignn_spmm_csr<<<nodeWaveBlocks, 256, 0, stream>>>(colptr, srowsC, svalsC, Gm, Out, N);
  };

  // ---- 0) build both CSRs (by col for SpMM, by row for power iteration) ----
  hipMemsetAsync(cntA, 0, (size_t)N * 4, stream);
  ignn_hist<<<eBlocks, 256, 0, stream>>>(ecols, cntA, E);
  ignn_exscan<<<1, 256, 0, stream>>>(cntA, colptr, N);
  hipMemcpyAsync(cntA, colptr, (size_t)N * 4, hipMemcpyDeviceToDevice, stream);
  ignn_csr_scatter<<<eBlocks, 256, 0, stream>>>(ecols, erows, edge_vals, cntA,
                                                srowsC, svalsC, E);
  hipMemsetAsync(cntA, 0, (size_t)N * 4, stream);
  ignn_hist<<<eBlocks, 256, 0, stream>>>(erows, cntA, E);
  ignn_exscan<<<1, 256, 0, stream>>>(cntA, rowptr, N);
  hipMemcpyAsync(cntA, rowptr, (size_t)N * 4, hipMemcpyDeviceToDevice, stream);
  ignn_csr_scatter<<<eBlocks, 256, 0, stream>>>(erows, ecols, edge_vals, cntA,
                                                scolsR, svalsR, E);

  // ---- 1) spectral radius: 100 power iterations (no atomics in SpMV) ----
  ignn_init_v<<<nBlocks, 256, 0, stream>>>(vvec, N);
  for (int it = 0; it < IG_POWER; ++it) {
    ignn_spmv_csr<<<nBlocks, 256, 0, stream>>>(rowptr, scolsR, svalsR, vvec, wvec, N);
    hipMemsetAsync(ss, 0, 4, stream);
    ignn_sumsq<<<nBlocks, 256, 0, stream>>>(wvec, ss, N);
    ignn_normalize<<<nBlocks, 256, 0, stream>>>(wvec, vvec, ss, rho, N);
  }

  // ---- 2) weight prep ----
  ignn_proj_norm_inf<<<IG_M, IG_M, 0, stream>>>(W1, rho, Wp1T);
  ignn_proj_norm_inf<<<IG_M, IG_M, 0, stream>>>(W2, rho, Wp2T);
  ignn_transpose<<<(IG_M * IG_P + 255) / 256, 256, 0, stream>>>(Omega1, OT1, IG_M, IG_P);
  ignn_transpose<<<(IG_M * IG_P + 255) / 256, 256, 0, stream>>>(V0_w, V0T, IG_M, IG_P);
  ignn_transpose<<<(IG_M * IG_M + 255) / 256, 256, 0, stream>>>(Omega2, OT2, IG_M, IG_M);
  ignn_transpose<<<(IG_M * IG_M + 255) / 256, 256, 0, stream>>>(V1_w, V1T, IG_M, IG_M);
  ignn_transpose<<<(int)(((size_t)IG_P * N + 255) / 256), 256, 0, stream>>>(
      features, featT, IG_P, N);

  // ---- 3) layer 1:  X = relu(Wp1 X A + Omega1 U A), U = features ----
  gemm(featT, OT1, G, IG_P);   // G = (Omega1 @ U)^T   [N,128]
  spmm(G, Bm);                 // Bm = b_Omega (node-major)
  hipMemcpyAsync(X, Bm, NM * 4, hipMemcpyDeviceToDevice, stream);
  for (int it = 0; it < IG_MITR; ++it) {
    gemm(X, Wp1T, G, IG_M);    // G = (Wp1 @ X)^T
    spmm(G, Acc);
    ignn_relu_add<<<eltBlocks, 256, 0, stream>>>(Acc, Bm, X, NM);
  }
  gemm(featT, V0T, Acc, IG_P); // U^T @ V0_w^T
  float* X2 = featT;           // featT no longer needed -> reuse as x2
  ignn_elu_combine<<<eltBlocks, 256, 0, stream>>>(X, Acc, V0_b, X2, NM);

  // ---- 4) layer 2 ----
  gemm(X2, OT2, G, IG_M);
  spmm(G, Bm);
  hipMemcpyAsync(X, Bm, NM * 4, hipMemcpyDeviceToDevice, stream);
  for (int it = 0; it < IG_MITR; ++it) {
    gemm(X, Wp2T, G, IG_M);
    spmm(G, Acc);
    ignn_relu_add<<<eltBlocks, 256, 0, stream>>>(Acc, Bm, X, NM);
  }

  // ---- 5) out = h2 + x2 @ V1_w^T + V1_b   [N,128] row-major ----
  gemm(X2, V1T, Acc, IG_M);
  ignn_final_combine<<<eltBlocks, 256, 0, stream>>>(X, Acc, V1_b, (float*)d_out, NM);
}